// LIFLayer_35124242547261
// MI455X (gfx1250) — compile-verified
//
#include <hip/hip_runtime.h>
#include <stdint.h>

// ---------------- problem constants (from reference module defaults) --------
#define T_STEPS 200
#define BATCH   64
#define NNEUR   4096
#define BN      (BATCH * NNEUR)   // 262144 elements per time plane
#define HALF    (BN / 2)          // 131072 : JAX threefry counter pairing offset
#define THREADS 256
#define VEC     4
#define BLOCKS  (HALF / (THREADS * VEC)) // 128

typedef float v4f __attribute__((ext_vector_type(4)));

// ---------------- Threefry-2x32 (20 rounds), matches JAX -------------------
__device__ __forceinline__ uint32_t tf_rotl(uint32_t x, uint32_t r) {
    return (x << r) | (x >> (32u - r));
}

__device__ __forceinline__ void tf2x32(uint32_t k0, uint32_t k1,
                                       uint32_t c0, uint32_t c1,
                                       uint32_t& o0, uint32_t& o1) {
    const uint32_t ks2 = k0 ^ k1 ^ 0x1BD11BDAu;
    uint32_t x0 = c0 + k0;
    uint32_t x1 = c1 + k1;
#define TF_R(r) { x0 += x1; x1 = tf_rotl(x1, r); x1 ^= x0; }
    TF_R(13) TF_R(15) TF_R(26) TF_R(6)
    x0 += k1;  x1 += ks2 + 1u;
    TF_R(17) TF_R(29) TF_R(16) TF_R(24)
    x0 += ks2; x1 += k0 + 2u;
    TF_R(13) TF_R(15) TF_R(26) TF_R(6)
    x0 += k0;  x1 += k1 + 3u;
    TF_R(17) TF_R(29) TF_R(16) TF_R(24)
    x0 += k1;  x1 += ks2 + 4u;
    TF_R(13) TF_R(15) TF_R(26) TF_R(6)
    x0 += ks2; x1 += k0 + 5u;
#undef TF_R
    o0 = x0; o1 = x1;
}

// bits -> uniform [0,1) (JAX mantissa trick)
__device__ __forceinline__ float tf_u01(uint32_t b) {
    return __uint_as_float((b >> 9) | 0x3F800000u) - 1.0f;
}

// bits -> standard normal (JAX: uniform(-1+ulp, 1) then sqrt(2)*erfinv)
__device__ __forceinline__ float tf_normal(uint32_t b) {
    const float lo = -0.99999994f;             // nextafter(-1, 0) in f32
    float f = tf_u01(b);
    float u = fmaxf(lo, fmaf(f, 1.0f - lo, lo));
    return 1.41421356237f * erfinvf(u);
}

// ---------------- per-step key derivation (JAX split tree) -----------------
// root = key(42) = (0, 42); keys = split(root, T); per step split -> (noise, spike)
__global__ void lif_keys_kernel(uint4* __restrict__ keys) {
    const int t = blockIdx.x * blockDim.x + threadIdx.x;
    if (t >= T_STEPS) return;
    const uint32_t RK0 = 0u, RK1 = 42u;
    const uint32_t j0 = 2u * (uint32_t)t;
    const uint32_t j1 = j0 + 1u;
    uint32_t b0, b1, o0, o1;
    // split(root, T): counters arange(2T) split in halves; bits[j] = out0 of
    // call (j, j+T) when j < T, else out1 of call (j-T, j).
    if (j0 < (uint32_t)T_STEPS) {
        tf2x32(RK0, RK1, j0, j0 + T_STEPS, o0, o1); b0 = o0;
        tf2x32(RK0, RK1, j1, j1 + T_STEPS, o0, o1); b1 = o0;
    } else {
        tf2x32(RK0, RK1, j0 - T_STEPS, j0, o0, o1); b0 = o1;
        tf2x32(RK0, RK1, j1 - T_STEPS, j1, o0, o1); b1 = o1;
    }
    // split(key_t, 2): counters [0,1,2,3] -> calls (0,2) and (1,3)
    uint32_t n0, n1, s0, s1;
    tf2x32(b0, b1, 0u, 2u, n0, s0);
    tf2x32(b0, b1, 1u, 3u, n1, s1);
    keys[t] = make_uint4(n0, n1, s0, s1); // k_noise=(n0,n1), k_spike=(s0,s1)
}

// ---------------- CDNA5 async global->LDS staging --------------------------
__device__ __forceinline__ void async_ld16(uint32_t lds_off, const float* gptr) {
    asm volatile("global_load_async_to_lds_b128 %0, %1, off"
                 :: "v"(lds_off), "v"((uint64_t)(uintptr_t)gptr)
                 : "memory");
}

// ---------------- LIF state update -----------------------------------------
__device__ __forceinline__ void lif_update(float x, float noise, float u01, float nm,
                                           float& V, float& Th, float& Ad, float& Sy,
                                           float& spk_out, float& v_out) {
    const float inv_tau = 0.05f;                       // DT / TAU
    float Ieff = fmaf(x, Sy, nm) - Ad;                 // I*syn + neuromod - adapt
    float v = fmaf(Ieff - V, inv_tau, V) + noise;
    float p = 1.0f / (1.0f + __expf(Th - v));          // sigmoid(v - Th)
    bool  s = u01 < p;
    float sf = s ? 1.0f : 0.0f;
    v = s ? 0.0f : v;                                  // V_RESET
    float th = s ? (Th + 0.1f) : fmaf(-0.1f, Th - 1.0f, Th);
    th = fminf(2.0f, fmaxf(0.5f, th));
    Ad = fmaf(Ad, 0.9f, 0.5f * sf);
    Sy = fmaf(Sy, 1.0f - 0.1f * sf, 0.0f) + 0.05f * (1.0f - Sy);
    V = v; Th = th;
    spk_out = sf; v_out = v;
}

__device__ __forceinline__ void unpack4(float* d, float4 a) {
    d[0] = a.x; d[1] = a.y; d[2] = a.z; d[3] = a.w;
}

// streaming (write-once) store: global_store_b128 th:TH_STORE_NT — keep the
// 420 MB output stream out of the 192 MB L2
__device__ __forceinline__ void store_nt4(float* p, const float* s) {
    v4f v = { s[0], s[1], s[2], s[3] };
    __builtin_nontemporal_store(v, (v4f*)p);
}

// ---------------- main scan kernel -----------------------------------------
__global__ __launch_bounds__(THREADS) void lif_scan_kernel(
    const float* __restrict__ in,
    const float* __restrict__ V0,  const float* __restrict__ Th0,
    const float* __restrict__ Ad0, const float* __restrict__ Sy0,
    const float* __restrict__ Nm0,
    const uint4* __restrict__ keys,
    float* __restrict__ spk, float* __restrict__ vlt)
{
    __shared__ float4 buf[2][2 * THREADS];   // double-buffered input tiles, 16 KB
    const int tid = threadIdx.x;
    const int jl  = blockIdx.x * (THREADS * VEC) + tid * VEC; // lower-half base
    const int ju  = jl + HALF;                                // paired upper-half

    // hoisted LDS byte offsets for the staging slots (lane-private, no barrier)
    const uint32_t lds_lo0 = (uint32_t)(uintptr_t)&buf[0][tid];
    const uint32_t lds_hi0 = (uint32_t)(uintptr_t)&buf[0][THREADS + tid];
    const uint32_t lds_lo1 = (uint32_t)(uintptr_t)&buf[1][tid];
    const uint32_t lds_hi1 = (uint32_t)(uintptr_t)&buf[1][THREADS + tid];

    // persistent per-neuron state in registers (lanes 0..3 lower, 4..7 upper)
    float V[8], Th[8], Ad[8], Sy[8], Nm[8];
    unpack4(V,      *(const float4*)(V0  + jl)); unpack4(V  + 4, *(const float4*)(V0  + ju));
    unpack4(Th,     *(const float4*)(Th0 + jl)); unpack4(Th + 4, *(const float4*)(Th0 + ju));
    unpack4(Ad,     *(const float4*)(Ad0 + jl)); unpack4(Ad + 4, *(const float4*)(Ad0 + ju));
    unpack4(Sy,     *(const float4*)(Sy0 + jl)); unpack4(Sy + 4, *(const float4*)(Sy0 + ju));
    unpack4(Nm,     *(const float4*)(Nm0 + jl)); unpack4(Nm + 4, *(const float4*)(Nm0 + ju));

    // prologue: stage t = 0
    async_ld16(lds_lo0, in + jl);
    async_ld16(lds_hi0, in + ju);

    for (int t = 0; t < T_STEPS; ++t) {
        const int cur = t & 1;
        if (t + 1 < T_STEPS) {                      // uniform branch
            const float* pn = in + (size_t)(t + 1) * BN;
            async_ld16(cur ? lds_lo0 : lds_lo1, pn + jl);
            async_ld16(cur ? lds_hi0 : lds_hi1, pn + ju);
            // current tile's 2 loads complete first (in-order): wait cnt <= 2
            asm volatile("s_wait_asynccnt 0x2" ::: "memory");
        } else {
            asm volatile("s_wait_asynccnt 0x0" ::: "memory");
        }

        float x[8];
        unpack4(x,     buf[cur][tid]);
        unpack4(x + 4, buf[cur][THREADS + tid]);

        const uint4 kk = keys[t];                   // uniform -> scalar load
        float nz[8], u[8];
#pragma unroll
        for (int i = 0; i < 4; ++i) {
            uint32_t o0, o1;
            // JAX counter pairing: element e (<HALF) pairs with e+HALF
            tf2x32(kk.x, kk.y, (uint32_t)(jl + i), (uint32_t)(ju + i), o0, o1);
            nz[i]     = 0.1f * tf_normal(o0);       // NOISE_STD
            nz[4 + i] = 0.1f * tf_normal(o1);
            tf2x32(kk.z, kk.w, (uint32_t)(jl + i), (uint32_t)(ju + i), o0, o1);
            u[i]     = tf_u01(o0);
            u[4 + i] = tf_u01(o1);
        }

        float so[8], vo[8];
#pragma unroll
        for (int i = 0; i < 8; ++i)
            lif_update(x[i], nz[i], u[i], Nm[i], V[i], Th[i], Ad[i], Sy[i], so[i], vo[i]);

        const size_t plane = (size_t)t * BN;
        store_nt4(spk + plane + jl, so);
        store_nt4(spk + plane + ju, so + 4);
        store_nt4(vlt + plane + jl, vo);
        store_nt4(vlt + plane + ju, vo + 4);
    }
}

// ---------------- launch ----------------------------------------------------
extern "C" void kernel_launch(void* const* d_in, const int* in_sizes, int n_in,
                              void* d_out, int out_size, void* d_ws, size_t ws_size,
                              hipStream_t stream) {
    const float* input = (const float*)d_in[0];
    const float* V0    = (const float*)d_in[1];
    const float* Th0   = (const float*)d_in[2];
    const float* Ad0   = (const float*)d_in[3];
    const float* Sy0   = (const float*)d_in[4];
    const float* Nm0   = (const float*)d_in[5];

    uint4* keys = (uint4*)d_ws;                       // 200 * 16 B = 3.2 KB
    float* spk  = (float*)d_out;                      // (T,B,N) spikes as 0/1 f32
    float* vlt  = spk + (size_t)T_STEPS * BN;         // (T,B,N) voltages

    lif_keys_kernel<<<1, 256, 0, stream>>>(keys);
    lif_scan_kernel<<<BLOCKS, THREADS, 0, stream>>>(input, V0, Th0, Ad0, Sy0, Nm0,
                                                    keys, spk, vlt);
}